// DCNv2_30958124270234
// MI455X (gfx1250) — compile-verified
//
#include <hip/hip_runtime.h>
#include <hip/hip_bf16.h>

typedef __attribute__((ext_vector_type(2))) float v2f;
typedef __attribute__((ext_vector_type(8))) float v8f;

#define Bn   4
#define Ci   256
#define Co   256
#define Hh   64
#define Ww   64
#define HW   4096          // 64*64
#define KK   9
#define Kdim 2304          // Ci*KK
#define NPIX 16384         // Bn*HW
#define KC   36            // k-chunk: 4 channels * 9 taps
#define NT   32            // pixels per workgroup
#define TSTR 40            // s_tile row stride (pad for banks, keeps 8B align)

// ---------------- Stage 1: offset/mask conv (om = conv3x3(feat, w_om) + b_om) ---------
__global__ void om_conv_kernel(const float* __restrict__ feat,
                               const float* __restrict__ w_om,
                               const float* __restrict__ b_om,
                               float* __restrict__ om) {
    int idx = blockIdx.x * blockDim.x + threadIdx.x;
    if (idx >= Bn * 27 * HW) return;
    int hw = idx & (HW - 1);
    int co = (idx >> 12) % 27;
    int b  = idx / (27 * HW);
    int h = hw >> 6, w = hw & 63;
    float acc = b_om[co];
    const float* fb = feat + (size_t)b * Ci * HW;
    const float* wb = w_om + (size_t)co * Ci * 9;
    for (int ci = 0; ci < Ci; ++ci) {
        const float* fp = fb + ci * HW;
        const float* wp = wb + ci * 9;
#pragma unroll
        for (int dy = 0; dy < 3; ++dy) {
            int yy = h + dy - 1;
            if (yy < 0 || yy > 63) continue;
#pragma unroll
            for (int dx = 0; dx < 3; ++dx) {
                int xx = w + dx - 1;
                if (xx < 0 || xx > 63) continue;
                acc = fmaf(fp[yy * 64 + xx], wp[dy * 3 + dx], acc);
            }
        }
    }
    om[idx] = acc;
}

// ---------------- Stage 2: fused deformable-sample + implicit GEMM (WMMA f32) ---------
// Workgroup: 256 threads (8 waves). Computes out tile [Co=256, 32 pixels].
// Each wave: 32 (M) x 32 (N) via four 16x16 f32 WMMA accumulators, K stepped by 4.
__global__ __launch_bounds__(256) void dcn_wmma_kernel(
        const float* __restrict__ x,
        const float* __restrict__ weight,   // [Co][Ci*9], k = c*9+kk
        const float* __restrict__ bias,
        const float* __restrict__ om,       // [B][27][64][64]
        float* __restrict__ out) {          // [B][Co][64][64], pre-BN
    __shared__ int   s_off[NT * KK * 4];    // clipped corner offsets within a channel plane
    __shared__ float s_wgt[NT * KK * 4];    // bilinear weight * valid * sigmoid(mask)
    __shared__ float s_tile[NT * TSTR];     // sampled tile, N-major [n][k_local]

    const int tid  = threadIdx.x;
    const int pix0 = blockIdx.x * NT;       // 32 consecutive pixels: same b, same row
    const int b    = pix0 >> 12;

    // ---- per-workgroup bilinear parameter precompute (32 pixels x 9 taps) ----
    for (int e = tid; e < NT * KK; e += 256) {
        int nl = e / KK, kk = e % KK;
        int pix = pix0 + nl;
        int hw = pix & (HW - 1);
        int h = hw >> 6, w = hw & 63;
        const float* omb = om + (size_t)b * 27 * HW;
        float o1 = omb[kk * HW + hw];
        float o2 = omb[(9 + kk) * HW + hw];
        float mm = omb[(18 + kk) * HW + hw];
        float mval = 1.0f / (1.0f + __expf(-mm));   // sigmoid(mask)
        float py = o1 + (float)(kk / 3 - 1) + (float)h;
        float px = o2 + (float)(kk % 3 - 1) + (float)w;
        float y0f = floorf(py), x0f = floorf(px);
        float dy = py - y0f, dx = px - x0f;
        int y0 = (int)y0f, x0 = (int)x0f;
#pragma unroll
        for (int cr = 0; cr < 4; ++cr) {
            int yy = y0 + (cr >> 1), xx = x0 + (cr & 1);
            bool v = (yy >= 0) && (yy < Hh) && (xx >= 0) && (xx < Ww);
            int yc = min(max(yy, 0), Hh - 1), xc = min(max(xx, 0), Ww - 1);
            float wgt = ((cr >> 1) ? dy : 1.0f - dy) * ((cr & 1) ? dx : 1.0f - dx);
            s_off[e * 4 + cr] = yc * Ww + xc;
            s_wgt[e * 4 + cr] = v ? wgt * mval : 0.0f;
        }
    }

    const int wave = tid >> 5, lane = tid & 31;
    const int half = lane >> 4, lm = lane & 15;
    const int mrow0 = wave * 32 + lm;

    const v8f zf = {0.f, 0.f, 0.f, 0.f, 0.f, 0.f, 0.f, 0.f};
    v8f acc[2][2];
    acc[0][0] = zf; acc[0][1] = zf; acc[1][0] = zf; acc[1][1] = zf;

    const float* xb = x + (size_t)b * Ci * HW;

    for (int chunk = 0; chunk < Kdim / KC; ++chunk) {
        const int c0 = chunk * 4;           // 4 input channels per chunk
        __syncthreads();                    // previous chunk's s_tile reads done
        // ---- build sampled B tile: s_tile[n][kl], kl = (c-c0)*9 + kk ----
        for (int i = tid; i < NT * KC; i += 256) {
            int nl = i / KC;
            int kl = i % KC;
            int e  = nl * KK + (kl % KK);
            const float* xp = xb + (c0 + kl / KK) * HW;
            float v = s_wgt[e * 4 + 0] * xp[s_off[e * 4 + 0]]
                    + s_wgt[e * 4 + 1] * xp[s_off[e * 4 + 1]]
                    + s_wgt[e * 4 + 2] * xp[s_off[e * 4 + 2]]
                    + s_wgt[e * 4 + 3] * xp[s_off[e * 4 + 3]];
            s_tile[nl * TSTR + kl] = v;
        }
        __syncthreads();

        const int kgbase = chunk * KC;
#pragma unroll
        for (int s = 0; s < KC / 4; ++s) {
            const int kl = s * 4 + 2 * half;      // fragment K = {kl, kl+1} this lane
            v2f afrag[2], bfrag[2];
#pragma unroll
            for (int mt = 0; mt < 2; ++mt) {
                const float* ap = weight + (size_t)(mrow0 + mt * 16) * Kdim + (kgbase + kl);
                afrag[mt] = *(const v2f*)ap;
            }
#pragma unroll
            for (int nt = 0; nt < 2; ++nt) {
                bfrag[nt] = *(const v2f*)&s_tile[(nt * 16 + lm) * TSTR + kl];
            }
#pragma unroll
            for (int mt = 0; mt < 2; ++mt) {
#pragma unroll
                for (int nt = 0; nt < 2; ++nt) {
                    acc[mt][nt] = __builtin_amdgcn_wmma_f32_16x16x4_f32(
                        false, afrag[mt], false, bfrag[nt],
                        (short)0, acc[mt][nt], false, false);
                }
            }
        }
    }

    // ---- epilogue: D layout -> out[b][co][hw] (+bias), pre-BN ----
#pragma unroll
    for (int mt = 0; mt < 2; ++mt) {
#pragma unroll
        for (int nt = 0; nt < 2; ++nt) {
            int pix = pix0 + nt * 16 + lm;
            int hw = pix & (HW - 1);
#pragma unroll
            for (int r = 0; r < 8; ++r) {
                int co = wave * 32 + mt * 16 + half * 8 + r;
                out[(size_t)b * (Co * HW) + (size_t)co * HW + hw] = acc[mt][nt][r] + bias[co];
            }
        }
    }
}

// ---------------- Stage 3: per-channel BN stats (mean, rsqrt(var+eps)) ---------------
__global__ void bn_stats_kernel(const float* __restrict__ out, float* __restrict__ stats) {
    int co = blockIdx.x;
    int tid = threadIdx.x;
    float s = 0.f, sq = 0.f;
    for (int i = tid; i < Bn * HW; i += 256) {
        int b = i >> 12, hw = i & (HW - 1);
        float v = out[(size_t)b * (Co * HW) + (size_t)co * HW + hw];
        s += v; sq += v * v;
    }
    __shared__ float rs[256], rq[256];
    rs[tid] = s; rq[tid] = sq;
    __syncthreads();
    for (int off = 128; off > 0; off >>= 1) {
        if (tid < off) { rs[tid] += rs[tid + off]; rq[tid] += rq[tid + off]; }
        __syncthreads();
    }
    if (tid == 0) {
        float mean = rs[0] * (1.0f / (Bn * HW));
        float var  = rq[0] * (1.0f / (Bn * HW)) - mean * mean;
        stats[co]      = mean;
        stats[Co + co] = rsqrtf(var + 1e-5f);
    }
}

// ---------------- Stage 4: apply BN in-place ------------------------------------------
__global__ void bn_apply_kernel(float* __restrict__ out, const float* __restrict__ stats,
                                const float* __restrict__ gamma, const float* __restrict__ beta) {
    int idx = blockIdx.x * blockDim.x + threadIdx.x;
    if (idx >= Bn * Co * HW) return;
    int co = (idx >> 12) & (Co - 1);
    float v = out[idx];
    out[idx] = gamma[co] * stats[Co + co] * (v - stats[co]) + beta[co];
}

extern "C" void kernel_launch(void* const* d_in, const int* in_sizes, int n_in,
                              void* d_out, int out_size, void* d_ws, size_t ws_size,
                              hipStream_t stream) {
    const float* x      = (const float*)d_in[0];
    const float* feat   = (const float*)d_in[1];
    const float* weight = (const float*)d_in[2];
    const float* bias   = (const float*)d_in[3];
    const float* w_om   = (const float*)d_in[4];
    const float* b_om   = (const float*)d_in[5];
    const float* gamma  = (const float*)d_in[6];
    const float* beta   = (const float*)d_in[7];
    float* out = (float*)d_out;

    float* om    = (float*)d_ws;                 // 4*27*4096 floats = 1.77 MB
    float* stats = om + (size_t)Bn * 27 * HW;    // 2*Co floats

    int om_total = Bn * 27 * HW;
    om_conv_kernel<<<(om_total + 255) / 256, 256, 0, stream>>>(feat, w_om, b_om, om);

    dcn_wmma_kernel<<<NPIX / NT, 256, 0, stream>>>(x, weight, bias, om, out);

    bn_stats_kernel<<<Co, 256, 0, stream>>>(out, stats);

    int total = Bn * Co * HW;
    bn_apply_kernel<<<(total + 255) / 256, 256, 0, stream>>>(out, stats, gamma, beta);
}